// AdaptivePrototypicalFewShotLearning_28432683500082
// MI455X (gfx1250) — compile-verified
//
#include <hip/hip_runtime.h>

typedef float v2f __attribute__((ext_vector_type(2)));
typedef float v4f __attribute__((ext_vector_type(4)));
typedef float v8f __attribute__((ext_vector_type(8)));

#define FEAT   512
#define HID    256
#define NCLS   20
#define NSUP   200
#define NQ     200000
#define LN_EPS 1e-5f

#define NWG    250
#define CHUNK  800        // 250 * 800 = 200000
#define QB     64
#define QSTR   513        // 513 % 64 == 1 -> conflict-free row-major A reads
#define CSTR   1028       // 1028 % 64 == 4
#define HSTR   513

__device__ __forceinline__ v8f wmma4(v2f a, v2f b, v8f c) {
  // V_WMMA_F32_16X16X4_F32 : D = A(16x4) x B(4x16) + C(16x16)
  return __builtin_amdgcn_wmma_f32_16x16x4_f32(false, a, false, b, (short)0, c,
                                               false, false);
}

// single-instruction XOR butterfly via ds_swizzle (group-of-32 mode:
// and_mask=0x1f, or_mask=0, xor_mask=M  ->  offset = 0x7C00 | M)
template <int M>
__device__ __forceinline__ float swzx(float v) {
  return __int_as_float(
      __builtin_amdgcn_ds_swizzle(__float_as_int(v), 0x7C00 | M));
}

__device__ __forceinline__ float rsum32(float v) {
  v += swzx<16>(v); v += swzx<8>(v); v += swzx<4>(v);
  v += swzx<2>(v);  v += swzx<1>(v);
  return v;
}
__device__ __forceinline__ float rsum16(float v) {
  v += swzx<8>(v); v += swzx<4>(v); v += swzx<2>(v); v += swzx<1>(v);
  return v;
}
__device__ __forceinline__ float rmin16(float v) {
  v = fminf(v, swzx<8>(v)); v = fminf(v, swzx<4>(v));
  v = fminf(v, swzx<2>(v)); v = fminf(v, swzx<1>(v));
  return v;
}

// ---------------------------------------------------------------- zero
__global__ void k_zero(float* p, int n) {
  int i = blockIdx.x * blockDim.x + threadIdx.x;
  if (i < n) p[i] = 0.0f;
}

// ---------------------------------------------------------------- prototypes
// support LN -> class means -> MLP -> LN. One block, 256 threads.
__global__ __launch_bounds__(256, 1) void k_protos(
    const float* __restrict__ sup, const int* __restrict__ lab,
    const float* __restrict__ lng, const float* __restrict__ lnb,
    const float* __restrict__ w1, const float* __restrict__ b1,
    const float* __restrict__ w2, const float* __restrict__ b2,
    float* __restrict__ protosLn, float* __restrict__ pp) {
  extern __shared__ float sm[];
  float* cme = sm;                 // 20*512
  float* cnt = cme + NCLS * FEAT;  // 32
  float* h   = cnt + 32;           // 20*256

  const int tid = threadIdx.x;
  const int lane = tid & 31;
  const int wv = tid >> 5;  // 8 waves

  for (int i = tid; i < NCLS * FEAT; i += 256) cme[i] = 0.0f;
  if (tid < 32) cnt[tid] = 0.0f;
  __syncthreads();

  // LN each support row, accumulate into class mean via LDS atomics
  for (int r = wv; r < NSUP; r += 8) {
    float x[16];
    float s = 0.0f;
#pragma unroll
    for (int j = 0; j < 16; ++j) { x[j] = sup[(size_t)r * FEAT + lane + 32 * j]; s += x[j]; }
    float mean = rsum32(s) * (1.0f / 512.0f);
    float vs = 0.0f;
#pragma unroll
    for (int j = 0; j < 16; ++j) { float d = x[j] - mean; vs += d * d; }
    float var = rsum32(vs) * (1.0f / 512.0f);
    float rq = rsqrtf(var + LN_EPS);
    int cls = lab[r];
#pragma unroll
    for (int j = 0; j < 16; ++j) {
      int c = lane + 32 * j;
      float y = (x[j] - mean) * rq * lng[c] + lnb[c];
      atomicAdd(&cme[cls * FEAT + c], y);
    }
    if (lane == 0) atomicAdd(&cnt[cls], 1.0f);
  }
  __syncthreads();
  for (int i = tid; i < NCLS * FEAT; i += 256) cme[i] /= fmaxf(cnt[i >> 9], 1.0f);
  __syncthreads();

  // h = relu(cme @ w1 + b1)   (20 x 256)
  for (int o = tid; o < NCLS * HID; o += 256) {
    int k = o >> 8, j = o & 255;
    float a = b1[j];
    for (int c = 0; c < FEAT; ++c) a += cme[k * FEAT + c] * w1[c * HID + j];
    h[k * HID + j] = fmaxf(a, 0.0f);
  }
  __syncthreads();
  // protos = h @ w2 + b2 -> overwrite cme
  for (int o = tid; o < NCLS * FEAT; o += 256) {
    int k = o >> 9, c = o & 511;
    float a = b2[c];
    for (int j = 0; j < HID; ++j) a += h[k * HID + j] * w2[j * FEAT + c];
    cme[o] = a;
  }
  __syncthreads();
  // LN rows, write padded protos (32 rows) + pp
  for (int r = wv; r < NCLS; r += 8) {
    float x[16];
    float s = 0.0f;
#pragma unroll
    for (int j = 0; j < 16; ++j) { x[j] = cme[r * FEAT + lane + 32 * j]; s += x[j]; }
    float mean = rsum32(s) * (1.0f / 512.0f);
    float vs = 0.0f;
#pragma unroll
    for (int j = 0; j < 16; ++j) { float d = x[j] - mean; vs += d * d; }
    float var = rsum32(vs) * (1.0f / 512.0f);
    float rq = rsqrtf(var + LN_EPS);
    float qq = 0.0f;
#pragma unroll
    for (int j = 0; j < 16; ++j) {
      int c = lane + 32 * j;
      float y = (x[j] - mean) * rq * lng[c] + lnb[c];
      protosLn[r * FEAT + c] = y;
      qq += y * y;
    }
    qq = rsum32(qq);
    if (lane == 0) pp[r] = qq;
  }
  for (int i = tid; i < 12 * FEAT; i += 256) protosLn[NCLS * FEAT + i] = 0.0f;
  if (tid < 12) pp[NCLS + tid] = 0.0f;
}

// ---------------------------------------------------------------- pass A
// per query: LN, WMMA dist vs protos, 3-temp softmax, WMMA-accumulate
// S_s = soft_s^T @ qn and wsum_s.
__global__ __launch_bounds__(512, 1) void k_passA(
    const float* __restrict__ qfeat, const float* __restrict__ lng,
    const float* __restrict__ lnb, const float* __restrict__ protosLn,
    const float* __restrict__ ppG, float* __restrict__ gwsum,
    float* __restrict__ gS) {
  extern __shared__ float sm[];
  float* qn   = sm;                      // 64 x QSTR
  float* pt   = qn + QB * QSTR;          // 512 x 32  (feature-major, 32 class cols)
  float* soft = pt + FEAT * 32;          // 3 x 64 x 32
  float* qqs  = soft + 3 * QB * 32;      // 64
  float* pps  = qqs + QB;                // 32
  float* gsh  = pps + 32;                // 512
  float* bsh  = gsh + FEAT;              // 512

  const int tid = threadIdx.x;
  const int lane = tid & 31;
  const int wv = tid >> 5;  // 16 waves
  const int hl = lane & 15;
  const bool lo = lane < 16;
  const int ks = lo ? 0 : 2;

  for (int i = tid; i < FEAT * 32; i += 512) pt[i] = protosLn[i];
  if (tid < 32) pps[tid] = ppG[tid];
  for (int i = tid; i < FEAT; i += 512) { gsh[i] = lng[i]; bsh[i] = lnb[i]; }
  __syncthreads();

  const int qbeg = blockIdx.x * CHUNK;
  int qend = qbeg + CHUNK; if (qend > NQ) qend = NQ;

  v8f zv = {0, 0, 0, 0, 0, 0, 0, 0};
  v8f acc[3][2][2];
#pragma unroll
  for (int s = 0; s < 3; ++s)
#pragma unroll
    for (int a = 0; a < 2; ++a)
#pragma unroll
      for (int b = 0; b < 2; ++b) acc[s][a][b] = zv;
  float wloc[3][2] = {{0, 0}, {0, 0}, {0, 0}};

  for (int q0 = qbeg; q0 < qend; q0 += QB) {
    // ---- load 64x512 block, non-temporal (streaming; bypass-hint caches)
    for (int i = tid; i < QB * 128; i += 512) {
      int r = i >> 7, c4 = (i & 127) << 2;
      int q = q0 + r;
      v4f v = {0, 0, 0, 0};
      if (q < qend)
        v = __builtin_nontemporal_load((const v4f*)(qfeat + (size_t)q * FEAT + c4));
      float* d = qn + r * QSTR + c4;
      d[0] = v.x; d[1] = v.y; d[2] = v.z; d[3] = v.w;
    }
    // prefetch next block while we compute on this one
    if (q0 + QB < qend) {
      const char* nx = (const char*)(qfeat + (size_t)(q0 + QB) * FEAT);
      __builtin_prefetch(nx + (size_t)tid * 256, 0, 0);
    }
    __syncthreads();
    // ---- LN (waves 0-3, 16 rows each)
    if (wv < 4) {
      for (int rr = 0; rr < 16; ++rr) {
        int r = wv * 16 + rr;
        float x[16];
        float s = 0.0f;
#pragma unroll
        for (int j = 0; j < 16; ++j) { x[j] = qn[r * QSTR + lane + 32 * j]; s += x[j]; }
        float mean = rsum32(s) * (1.0f / 512.0f);
        float vs = 0.0f;
#pragma unroll
        for (int j = 0; j < 16; ++j) { float d = x[j] - mean; vs += d * d; }
        float var = rsum32(vs) * (1.0f / 512.0f);
        float rq = rsqrtf(var + LN_EPS);
        float qq = 0.0f;
#pragma unroll
        for (int j = 0; j < 16; ++j) {
          int c = lane + 32 * j;
          float y = (x[j] - mean) * rq * gsh[c] + bsh[c];
          qn[r * QSTR + c] = y;
          qq += y * y;
        }
        qq = rsum32(qq);
        if (lane == 0) qqs[r] = qq;
      }
    }
    __syncthreads();
    // ---- dist (WMMA) + 3-temperature softmax (waves 0-3, one 16-query tile each)
    if (wv < 4) {
      const int arow = wv * 16 + hl;
      v8f c1 = zv, c2 = zv;
#pragma unroll 4
      for (int kk = 0; kk < FEAT; kk += 4) {
        int ka = kk + ks;
        v2f a;  a.x = qn[arow * QSTR + ka];     a.y = qn[arow * QSTR + ka + 1];
        v2f bA; bA.x = pt[ka * 32 + hl];        bA.y = pt[(ka + 1) * 32 + hl];
        v2f bB; bB.x = pt[ka * 32 + 16 + hl];   bB.y = pt[(ka + 1) * 32 + 16 + hl];
        c1 = wmma4(a, bA, c1);
        c2 = wmma4(a, bB, c2);
      }
      const int ca = hl, cb = hl + 16;
      const bool vb = (cb < NCLS);
      const float ppA = pps[ca], ppB = pps[cb];
#pragma unroll
      for (int v = 0; v < 8; ++v) {
        int m = lo ? v : v + 8;
        int r = wv * 16 + m;
        bool vq = (q0 + r) < qend;
        float qq = qqs[r];
        float d1 = fmaxf(qq + ppA - 2.0f * c1[v], 0.0f);
        float d2 = fmaxf(qq + ppB - 2.0f * c2[v], 0.0f);
        float dm = rmin16(vb ? fminf(d1, d2) : d1);
#pragma unroll
        for (int s = 0; s < 3; ++s) {
          float invT = 1.0f / (float)(s + 1);
          float e1 = vq ? __expf((dm - d1) * invT) : 0.0f;
          float e2 = (vq && vb) ? __expf((dm - d2) * invT) : 0.0f;
          float sum = rsum16(e1 + e2);
          float inv = vq ? (1.0f / sum) : 0.0f;
          float s1 = e1 * inv, s2 = e2 * inv;
          soft[(s * QB + r) * 32 + ca] = s1;
          soft[(s * QB + r) * 32 + cb] = s2;
          wloc[s][0] += s1;
          wloc[s][1] += s2;
        }
      }
    }
    __syncthreads();
    // ---- accumulate S_s += soft_s^T @ qn (all 16 waves; wave owns feat tiles wv, wv+16)
    {
      const int n0 = wv * 16 + hl;
      const int n1 = (wv + 16) * 16 + hl;
#pragma unroll 1
      for (int kk = 0; kk < QB; kk += 4) {
        int qa = kk + ks, qb = qa + 1;
        v2f bA; bA.x = qn[qa * QSTR + n0]; bA.y = qn[qb * QSTR + n0];
        v2f bB; bB.x = qn[qa * QSTR + n1]; bB.y = qn[qb * QSTR + n1];
#pragma unroll
        for (int s = 0; s < 3; ++s) {
          v2f a0; a0.x = soft[(s * QB + qa) * 32 + hl];      a0.y = soft[(s * QB + qb) * 32 + hl];
          v2f a1; a1.x = soft[(s * QB + qa) * 32 + 16 + hl]; a1.y = soft[(s * QB + qb) * 32 + 16 + hl];
          acc[s][0][0] = wmma4(a0, bA, acc[s][0][0]);
          acc[s][0][1] = wmma4(a0, bB, acc[s][0][1]);
          acc[s][1][0] = wmma4(a1, bA, acc[s][1][0]);
          acc[s][1][1] = wmma4(a1, bB, acc[s][1][1]);
        }
      }
    }
    __syncthreads();
  }
  // ---- flush wsum
  if (wv < 4) {
#pragma unroll
    for (int s = 0; s < 3; ++s) {
      atomicAdd(&gwsum[s * 32 + hl], wloc[s][0]);
      if (hl + 16 < NCLS) atomicAdd(&gwsum[s * 32 + 16 + hl], wloc[s][1]);
    }
  }
  // ---- flush S accumulators
#pragma unroll
  for (int s = 0; s < 3; ++s)
#pragma unroll
    for (int mt = 0; mt < 2; ++mt)
#pragma unroll
      for (int j = 0; j < 2; ++j) {
        int nt = (j == 0) ? wv : wv + 16;
        int feat = nt * 16 + hl;
#pragma unroll
        for (int v = 0; v < 8; ++v) {
          int cls = mt * 16 + v + (lo ? 0 : 8);
          if (cls < NCLS)
            atomicAdd(&gS[((s * NCLS) + cls) * FEAT + feat], acc[s][mt][j][v]);
        }
      }
}

// ---------------------------------------------------------------- refine
// 3 refinement steps (WMMA GEMMs, weights streamed from global), final LN.
__global__ __launch_bounds__(512, 1) void k_refine(
    const float* __restrict__ protosLn, const float* __restrict__ gwsum,
    const float* __restrict__ gS, const float* __restrict__ w1,
    const float* __restrict__ b1, const float* __restrict__ w2,
    const float* __restrict__ b2, const float* __restrict__ lng,
    const float* __restrict__ lnb, float* __restrict__ rfl,
    float* __restrict__ pp2) {
  extern __shared__ float sm[];
  float* cat = sm;                 // 32 x CSTR  ([:, :512] holds refined)
  float* h   = cat + 32 * CSTR;    // 32 x HSTR

  const int tid = threadIdx.x;
  const int lane = tid & 31;
  const int wv = tid >> 5;  // 16 waves
  const int hl = lane & 15;
  const bool lo = lane < 16;
  const int ks = lo ? 0 : 2;
  const int colA = wv * 16 + hl;
  const int colB = (wv + 16) * 16 + hl;
  v8f zv = {0, 0, 0, 0, 0, 0, 0, 0};

  for (int i = tid; i < 32 * CSTR; i += 512) cat[i] = 0.0f;
  __syncthreads();
  for (int i = tid; i < NCLS * FEAT; i += 512)
    cat[(i >> 9) * CSTR + (i & 511)] = protosLn[i];
  __syncthreads();

  for (int s = 0; s < 3; ++s) {
    // wmean into cat[:, 512:1024]
    for (int i = tid; i < NCLS * FEAT; i += 512) {
      int k = i >> 9, c = i & 511;
      float wsv = fmaxf(gwsum[s * 32 + k], 1e-6f);
      cat[k * CSTR + FEAT + c] = gS[(s * NCLS + k) * FEAT + c] / wsv;
    }
    __syncthreads();
    // GEMM1: h = relu(cat @ w1 + b1)    M=32(pad) N=512 K=1024
    v8f h00 = zv, h01 = zv, h10 = zv, h11 = zv;
#pragma unroll 2
    for (int kk = 0; kk < 1024; kk += 4) {
      int ka = kk + ks, kb = ka + 1;
      v2f a0; a0.x = cat[hl * CSTR + ka];        a0.y = cat[hl * CSTR + kb];
      v2f a1; a1.x = cat[(16 + hl) * CSTR + ka]; a1.y = cat[(16 + hl) * CSTR + kb];
      v2f bA; bA.x = w1[ka * FEAT + colA];       bA.y = w1[kb * FEAT + colA];
      v2f bB; bB.x = w1[ka * FEAT + colB];       bB.y = w1[kb * FEAT + colB];
      h00 = wmma4(a0, bA, h00);
      h01 = wmma4(a0, bB, h01);
      h10 = wmma4(a1, bA, h10);
      h11 = wmma4(a1, bB, h11);
    }
    float b1A = b1[colA], b1B = b1[colB];
#pragma unroll
    for (int v = 0; v < 8; ++v) {
      int m0 = v + (lo ? 0 : 8);
      h[m0 * HSTR + colA] = fmaxf(h00[v] + b1A, 0.0f);
      h[m0 * HSTR + colB] = fmaxf(h01[v] + b1B, 0.0f);
      int m1 = 16 + m0;
      h[m1 * HSTR + colA] = fmaxf(h10[v] + b1A, 0.0f);
      h[m1 * HSTR + colB] = fmaxf(h11[v] + b1B, 0.0f);
    }
    __syncthreads();
    // GEMM2: delta = h @ w2; refined += 0.1*(delta + b2)
    v8f d00 = zv, d01 = zv, d10 = zv, d11 = zv;
#pragma unroll 2
    for (int kk = 0; kk < FEAT; kk += 4) {
      int ka = kk + ks, kb = ka + 1;
      v2f a0; a0.x = h[hl * HSTR + ka];        a0.y = h[hl * HSTR + kb];
      v2f a1; a1.x = h[(16 + hl) * HSTR + ka]; a1.y = h[(16 + hl) * HSTR + kb];
      v2f bA; bA.x = w2[ka * FEAT + colA];     bA.y = w2[kb * FEAT + colA];
      v2f bB; bB.x = w2[ka * FEAT + colB];     bB.y = w2[kb * FEAT + colB];
      d00 = wmma4(a0, bA, d00);
      d01 = wmma4(a0, bB, d01);
      d10 = wmma4(a1, bA, d10);
      d11 = wmma4(a1, bB, d11);
    }
    float b2A = b2[colA], b2B = b2[colB];
#pragma unroll
    for (int v = 0; v < 8; ++v) {
      int m0 = v + (lo ? 0 : 8);
      cat[m0 * CSTR + colA] += 0.1f * (d00[v] + b2A);
      cat[m0 * CSTR + colB] += 0.1f * (d01[v] + b2B);
      int m1 = 16 + m0;
      if (m1 < NCLS) {
        cat[m1 * CSTR + colA] += 0.1f * (d10[v] + b2A);
        cat[m1 * CSTR + colB] += 0.1f * (d11[v] + b2B);
      }
    }
    __syncthreads();
  }
  // final LN of refined rows -> rfl (padded 32 rows) + pp2
  for (int r = wv; r < NCLS; r += 16) {
    float x[16];
    float s = 0.0f;
#pragma unroll
    for (int j = 0; j < 16; ++j) { x[j] = cat[r * CSTR + lane + 32 * j]; s += x[j]; }
    float mean = rsum32(s) * (1.0f / 512.0f);
    float vs = 0.0f;
#pragma unroll
    for (int j = 0; j < 16; ++j) { float d = x[j] - mean; vs += d * d; }
    float var = rsum32(vs) * (1.0f / 512.0f);
    float rq = rsqrtf(var + LN_EPS);
    float qq = 0.0f;
#pragma unroll
    for (int j = 0; j < 16; ++j) {
      int c = lane + 32 * j;
      float y = (x[j] - mean) * rq * lng[c] + lnb[c];
      rfl[r * FEAT + c] = y;
      qq += y * y;
    }
    qq = rsum32(qq);
    if (lane == 0) pp2[r] = qq;
  }
  for (int i = tid; i < 12 * FEAT; i += 512) rfl[NCLS * FEAT + i] = 0.0f;
  if (tid < 12) pp2[NCLS + tid] = 0.0f;
}

// ---------------------------------------------------------------- logits
__global__ __launch_bounds__(512, 1) void k_logits(
    const float* __restrict__ qfeat, const float* __restrict__ lng,
    const float* __restrict__ lnb, const float* __restrict__ rfl,
    const float* __restrict__ pp2, const float* __restrict__ dtemp,
    float* __restrict__ out) {
  extern __shared__ float sm[];
  float* qn  = sm;                  // 64 x QSTR
  float* pt  = qn + QB * QSTR;      // 512 x 32
  float* qqs = pt + FEAT * 32;      // 64
  float* pps = qqs + QB;            // 32
  float* gsh = pps + 32;            // 512
  float* bsh = gsh + FEAT;          // 512

  const int tid = threadIdx.x;
  const int lane = tid & 31;
  const int wv = tid >> 5;
  const int hl = lane & 15;
  const bool lo = lane < 16;
  const int ks = lo ? 0 : 2;
  const float temp = dtemp[0];

  for (int i = tid; i < FEAT * 32; i += 512) pt[i] = rfl[i];
  if (tid < 32) pps[tid] = pp2[tid];
  for (int i = tid; i < FEAT; i += 512) { gsh[i] = lng[i]; bsh[i] = lnb[i]; }
  __syncthreads();

  const int qbeg = blockIdx.x * CHUNK;
  int qend = qbeg + CHUNK; if (qend > NQ) qend = NQ;
  v8f zv = {0, 0, 0, 0, 0, 0, 0, 0};

  for (int q0 = qbeg; q0 < qend; q0 += QB) {
    for (int i = tid; i < QB * 128; i += 512) {
      int r = i >> 7, c4 = (i & 127) << 2;
      int q = q0 + r;
      v4f v = {0, 0, 0, 0};
      if (q < qend)
        v = __builtin_nontemporal_load((const v4f*)(qfeat + (size_t)q * FEAT + c4));
      float* d = qn + r * QSTR + c4;
      d[0] = v.x; d[1] = v.y; d[2] = v.z; d[3] = v.w;
    }
    if (q0 + QB < qend) {
      const char* nx = (const char*)(qfeat + (size_t)(q0 + QB) * FEAT);
      __builtin_prefetch(nx + (size_t)tid * 256, 0, 0);
    }
    __syncthreads();
    if (wv < 4) {
      for (int rr = 0; rr < 16; ++rr) {
        int r = wv * 16 + rr;
        float x[16];
        float s = 0.0f;
#pragma unroll
        for (int j = 0; j < 16; ++j) { x[j] = qn[r * QSTR + lane + 32 * j]; s += x[j]; }
        float mean = rsum32(s) * (1.0f / 512.0f);
        float vs = 0.0f;
#pragma unroll
        for (int j = 0; j < 16; ++j) { float d = x[j] - mean; vs += d * d; }
        float var = rsum32(vs) * (1.0f / 512.0f);
        float rq = rsqrtf(var + LN_EPS);
        float qq = 0.0f;
#pragma unroll
        for (int j = 0; j < 16; ++j) {
          int c = lane + 32 * j;
          float y = (x[j] - mean) * rq * gsh[c] + bsh[c];
          qn[r * QSTR + c] = y;
          qq += y * y;
        }
        qq = rsum32(qq);
        if (lane == 0) qqs[r] = qq;
      }
    }
    __syncthreads();
    if (wv < 4) {
      const int arow = wv * 16 + hl;
      v8f c1 = zv, c2 = zv;
#pragma unroll 4
      for (int kk = 0; kk < FEAT; kk += 4) {
        int ka = kk + ks;
        v2f a;  a.x = qn[arow * QSTR + ka];   a.y = qn[arow * QSTR + ka + 1];
        v2f bA; bA.x = pt[ka * 32 + hl];      bA.y = pt[(ka + 1) * 32 + hl];
        v2f bB; bB.x = pt[ka * 32 + 16 + hl]; bB.y = pt[(ka + 1) * 32 + 16 + hl];
        c1 = wmma4(a, bA, c1);
        c2 = wmma4(a, bB, c2);
      }
      const int ca = hl, cb = hl + 16;
      const float ppA = pps[ca], ppB = pps[cb];
#pragma unroll
      for (int v = 0; v < 8; ++v) {
        int m = lo ? v : v + 8;
        int r = wv * 16 + m;
        int q = q0 + r;
        if (q < qend) {
          float qq = qqs[r];
          float d1 = fmaxf(qq + ppA - 2.0f * c1[v], 0.0f);
          __builtin_nontemporal_store(-d1 * temp, &out[(size_t)q * NCLS + ca]);
          if (cb < NCLS) {
            float d2 = fmaxf(qq + ppB - 2.0f * c2[v], 0.0f);
            __builtin_nontemporal_store(-d2 * temp, &out[(size_t)q * NCLS + cb]);
          }
        }
      }
    }
    __syncthreads();
  }
}

// ---------------------------------------------------------------- launcher
extern "C" void kernel_launch(void* const* d_in, const int* in_sizes, int n_in,
                              void* d_out, int out_size, void* d_ws,
                              size_t ws_size, hipStream_t stream) {
  const float* sup   = (const float*)d_in[0];
  const int*   lab   = (const int*)d_in[1];
  const float* qry   = (const float*)d_in[2];
  const float* lng   = (const float*)d_in[3];
  const float* lnb   = (const float*)d_in[4];
  const float* pw1   = (const float*)d_in[5];
  const float* pb1   = (const float*)d_in[6];
  const float* pw2   = (const float*)d_in[7];
  const float* pb2   = (const float*)d_in[8];
  const float* rw1   = (const float*)d_in[9];
  const float* rb1   = (const float*)d_in[10];
  const float* rw2   = (const float*)d_in[11];
  const float* rb2   = (const float*)d_in[12];
  const float* dtemp = (const float*)d_in[13];
  float* out = (float*)d_out;
  float* ws  = (float*)d_ws;

  float* protosLn = ws;           // 32*512
  float* pp       = ws + 16384;   // 32
  float* gwsum    = ws + 16416;   // 3*32
  float* gS       = ws + 16512;   // 3*20*512
  float* rfl      = ws + 47232;   // 32*512
  float* pp2      = ws + 63616;   // 32

  const size_t sm1 = (size_t)(NCLS * FEAT + 32 + NCLS * HID) * 4;           // ~61.5 KB
  const size_t sm2 = (size_t)(QB * QSTR + FEAT * 32 + 3 * QB * 32 + QB + 32 + 2 * FEAT) * 4;  // ~226 KB
  const size_t sm3 = (size_t)(32 * CSTR + 32 * HSTR) * 4;                    // ~197 KB
  const size_t sm4 = (size_t)(QB * QSTR + FEAT * 32 + QB + 32 + 2 * FEAT) * 4;  // ~201 KB

  const int nz = 3 * 32 + 3 * NCLS * FEAT;  // gwsum + gS
  k_zero<<<(nz + 255) / 256, 256, 0, stream>>>(gwsum, nz);
  k_protos<<<1, 256, sm1, stream>>>(sup, lab, lng, lnb, pw1, pb1, pw2, pb2,
                                    protosLn, pp);
  k_passA<<<NWG, 512, sm2, stream>>>(qry, lng, lnb, protosLn, pp, gwsum, gS);
  k_refine<<<1, 512, sm3, stream>>>(protosLn, gwsum, gS, rw1, rb1, rw2, rb2,
                                    lng, lnb, rfl, pp2);
  k_logits<<<NWG, 512, sm4, stream>>>(qry, lng, lnb, rfl, pp2, dtemp, out);
}